// SparseMCFModel_54743653154937
// MI455X (gfx1250) — compile-verified
//
#include <hip/hip_runtime.h>
#include <math.h>
#include <stdint.h>

#define NNODES 20000
#define NEDGES 200000
#define DIM 64
#define HEADS 4
#define GLAYERS 2
#define FLOWITERS 10
#define GTPB 256   // 8 waves per block

typedef __attribute__((ext_vector_type(2))) float v2f;
typedef __attribute__((ext_vector_type(8))) float v8f;

// ---------------------------------------------------------------------------
// V_WMMA_F32_16X16X4_F32 wrapper (exact f32 GEMM path; 8-arg VOP3P pattern)
// ---------------------------------------------------------------------------
__device__ __forceinline__ v8f wmma4f32(v2f a, v2f b, v8f c) {
  return __builtin_amdgcn_wmma_f32_16x16x4_f32(
      /*neg_a=*/false, a, /*neg_b=*/false, b,
      /*c_mod=*/(short)0, c, /*reuse_a=*/false, /*reuse_b=*/false);
}

// ---------------------------------------------------------------------------
// Batched WMMA GEMM, K == 64 fixed, Ncol handled in 64-wide column panels
// (blockIdx.z). C[b][:, panel] = A[b](Mx64, lda) * B[b](64xNcol, ldb) + bias.
//
// - B panel (64x64 f32 = 16 KB) is staged into LDS with the CDNA5 async
//   datapath: global_load_async_to_lds_b128 (GVS mode) + s_wait_asynccnt,
//   then a workgroup barrier. All threads stage; guard comes after.
// - Each wave owns a 16-row x 64-col strip: one A fragment per K-step feeds
//   4 WMMAs (4 accumulators), B fragments come from LDS.
// - Fragment layouts per ISA 7.12.2: A 16x4 (lane<16 -> K={k,k+1}, lane>=16 ->
//   K={k+2,k+3}); B 4x16 mirrored by column; C/D row = v + 8*(lane>=16).
// - Strip guard is wave-uniform => EXEC all-1s for every executed WMMA.
// ---------------------------------------------------------------------------
__global__ void gemm_wmma_kernel(const float* __restrict__ A, int lda, long sA,
                                 const float* __restrict__ B, int ldb, long sB,
                                 float* __restrict__ C, int ldc, long sC,
                                 int M, const float* __restrict__ bias) {
  __shared__ float Bs[64 * 64];
  const int tid  = threadIdx.x;
  const int lane = tid & 31;
  const int wid  = tid >> 5;
  const int half = lane >> 4;   // 0: lanes 0-15, 1: lanes 16-31
  const int l16  = lane & 15;
  const int batch = blockIdx.y;
  const int col0  = blockIdx.z * 64;   // 64-wide column panel
  A += (long)batch * sA;
  B += (long)batch * sB + col0;
  C += (long)batch * sC + col0;

  // ---- async-stage B panel (64 rows x 64 cols, row stride ldb) into LDS ----
  {
    const uint64_t bbase = (uint64_t)(uintptr_t)B;
#pragma unroll
    for (int i = 0; i < 4; ++i) {
      int idx = tid * 4 + i;               // 1024 chunks of 16 B
      int kk  = idx >> 4;                  // B row   (K index) 0..63
      int c4  = (idx & 15) << 2;           // B col   0,4,...,60
      unsigned goff  = (unsigned)((kk * ldb + c4) * 4);
      unsigned laddr = (unsigned)(uintptr_t)(&Bs[kk * 64 + c4]);
      asm volatile("global_load_async_to_lds_b128 %0, %1, %2 offset:0"
                   :
                   : "v"(laddr), "v"(goff), "s"(bbase)
                   : "memory");
    }
    asm volatile("s_wait_asynccnt 0x0" ::: "memory");
  }
  __syncthreads();

  const int strip = blockIdx.x * (GTPB / 32) + wid;  // 16-row output strip
  if (strip < (M >> 4)) {
    const int row0 = strip << 4;
    const float* Arow = A + (long)(row0 + l16) * lda + 2 * half;
    v8f acc0 = {}, acc1 = {}, acc2 = {}, acc3 = {};
    for (int k = 0; k < 64; k += 4) {
      v2f a;
      a.x = Arow[k];
      a.y = Arow[k + 1];
      const float* brow0 = &Bs[(k + 2 * half) * 64 + l16];
      const float* brow1 = brow0 + 64;
      v2f b0, b1, b2, b3;
      b0.x = brow0[0];  b0.y = brow1[0];
      b1.x = brow0[16]; b1.y = brow1[16];
      b2.x = brow0[32]; b2.y = brow1[32];
      b3.x = brow0[48]; b3.y = brow1[48];
      acc0 = wmma4f32(a, b0, acc0);
      acc1 = wmma4f32(a, b1, acc1);
      acc2 = wmma4f32(a, b2, acc2);
      acc3 = wmma4f32(a, b3, acc3);
    }
    float bv0 = 0.f, bv1 = 0.f, bv2 = 0.f, bv3 = 0.f;
    if (bias) {
      bv0 = bias[col0 + l16];
      bv1 = bias[col0 + 16 + l16];
      bv2 = bias[col0 + 32 + l16];
      bv3 = bias[col0 + 48 + l16];
    }
#pragma unroll
    for (int v = 0; v < 8; ++v) {
      int rr = row0 + v + 8 * half;
      float* crow = C + (long)rr * ldc + l16;
      crow[0]  = acc0[v] + bv0;
      crow[16] = acc1[v] + bv1;
      crow[32] = acc2[v] + bv2;
      crow[48] = acc3[v] + bv3;
    }
  }
}

// ---------------------------------------------------------------------------
// s_src[h,n] = f[h,n,:].a_src[h,:]   s_dst likewise. One wave per (h,n) row.
// ---------------------------------------------------------------------------
__global__ void sdot_kernel(const float* __restrict__ f,
                            const float* __restrict__ a_src,
                            const float* __restrict__ a_dst,
                            float* __restrict__ s_src,
                            float* __restrict__ s_dst) {
  const int lane = threadIdx.x & 31;
  const int wid  = threadIdx.x >> 5;
  int idx = blockIdx.x * (blockDim.x >> 5) + wid;
  if (idx >= HEADS * NNODES) return;
  const int hh = idx / NNODES;
  const float* row = f + (long)idx * DIM;
  float x0 = row[lane], x1 = row[lane + 32];
  float ps = x0 * a_src[hh * DIM + lane] + x1 * a_src[hh * DIM + lane + 32];
  float pd = x0 * a_dst[hh * DIM + lane] + x1 * a_dst[hh * DIM + lane + 32];
  for (int off = 16; off > 0; off >>= 1) {
    ps += __shfl_down(ps, off, 32);
    pd += __shfl_down(pd, off, 32);
  }
  if (lane == 0) { s_src[idx] = ps; s_dst[idx] = pd; }
}

__global__ void fill_kernel(float* __restrict__ p, float v, int n) {
  int i = blockIdx.x * blockDim.x + threadIdx.x;
  if (i < n) p[i] = v;
}

__device__ __forceinline__ void atomicMaxF(float* addr, float v) {
  if (v >= 0.0f) atomicMax((int*)addr, __float_as_int(v));
  else           atomicMin((unsigned int*)addr, __float_as_uint(v));
}

// logits + segment-max (per head), leaky_relu slope 0.2
__global__ void edge_logit_kernel(const float* __restrict__ s_src,
                                  const float* __restrict__ s_dst,
                                  const int* __restrict__ erow,
                                  const int* __restrict__ ecol,
                                  float* __restrict__ elog,
                                  float* __restrict__ segmax) {
  int e = blockIdx.x * blockDim.x + threadIdx.x;
  if (e >= NEDGES) return;
  int r = erow[e], c = ecol[e];
#pragma unroll
  for (int hh = 0; hh < HEADS; ++hh) {
    float l = s_src[hh * NNODES + r] + s_dst[hh * NNODES + c];
    l = (l > 0.0f) ? l : 0.2f * l;
    elog[(long)hh * NEDGES + e] = l;
    atomicMaxF(&segmax[hh * NNODES + r], l);
  }
}

// exp(logit - segmax) in place + segment-sum
__global__ void edge_exp_kernel(const int* __restrict__ erow,
                                float* __restrict__ elog,
                                const float* __restrict__ segmax,
                                float* __restrict__ segsum) {
  int e = blockIdx.x * blockDim.x + threadIdx.x;
  if (e >= NEDGES) return;
  int r = erow[e];
#pragma unroll
  for (int hh = 0; hh < HEADS; ++hh) {
    float ev = expf(elog[(long)hh * NEDGES + e] - segmax[hh * NNODES + r]);
    elog[(long)hh * NEDGES + e] = ev;
    atomicAdd(&segsum[hh * NNODES + r], ev);
  }
}

// alpha * f[:,col,:] scattered into single [N,D] accumulator (head-mean folded
// in as 1/H). One wave per edge: lane handles dims {lane, lane+32}.
__global__ void scatter_kernel(const float* __restrict__ f,
                               const float* __restrict__ elog,
                               const float* __restrict__ segsum,
                               const int* __restrict__ erow,
                               const int* __restrict__ ecol,
                               float* __restrict__ gat) {
  const int lane = threadIdx.x & 31;
  const int wid  = threadIdx.x >> 5;
  int e = blockIdx.x * (blockDim.x >> 5) + wid;
  if (e >= NEDGES) return;
  int r = erow[e], c = ecol[e];
  float* dst = gat + (long)r * DIM;
#pragma unroll
  for (int hh = 0; hh < HEADS; ++hh) {
    float alpha = 0.25f * elog[(long)hh * NEDGES + e] / segsum[hh * NNODES + r];
    const float* src = f + ((long)hh * NNODES + c) * DIM;
    atomicAdd(&dst[lane],      alpha * src[lane]);
    atomicAdd(&dst[lane + 32], alpha * src[lane + 32]);
  }
}

// z, r gates; emits z and r*h.  G1=[N,192] (gat@Wx+b), G2=[N,128] (h@Wh[:, :2D])
__global__ void gru1_kernel(const float* __restrict__ G1,
                            const float* __restrict__ G2,
                            const float* __restrict__ h,
                            float* __restrict__ zbuf,
                            float* __restrict__ rh) {
  int i = blockIdx.x * blockDim.x + threadIdx.x;
  if (i >= NNODES * DIM) return;
  int n = i >> 6, j = i & 63;
  long o1 = (long)n * 192 + j;
  long o2 = (long)n * 128 + j;
  float z = 1.0f / (1.0f + expf(-(G1[o1]      + G2[o2])));
  float r = 1.0f / (1.0f + expf(-(G1[o1 + 64] + G2[o2 + 64])));
  zbuf[i] = z;
  rh[i]   = r * h[i];
}

// h = z*h + (1-z)*relu(G1[:,2D:] + C3)
__global__ void gru2_kernel(const float* __restrict__ G1,
                            const float* __restrict__ C3,
                            const float* __restrict__ zbuf,
                            float* __restrict__ h) {
  int i = blockIdx.x * blockDim.x + threadIdx.x;
  if (i >= NNODES * DIM) return;
  int n = i >> 6, j = i & 63;
  float cand = G1[(long)n * 192 + 128 + j] + C3[i];
  cand = cand > 0.0f ? cand : 0.0f;
  float z = zbuf[i];
  h[i] = z * h[i] + (1.0f - z) * cand;
}

// ---------------------------------------------------------------------------
// seg_softmax(pred_w[edge_row], edge_row) is exactly 1/degree(row): all edges
// in a segment share one logit => exp(logit - segmax) == exp(0) == 1 exactly.
// ---------------------------------------------------------------------------
__global__ void degree_kernel(const int* __restrict__ erow, float* __restrict__ deg) {
  int e = blockIdx.x * blockDim.x + threadIdx.x;
  if (e >= NEDGES) return;
  atomicAdd(&deg[erow[e]], 1.0f);
}
__global__ void invdeg_kernel(const int* __restrict__ erow,
                              const float* __restrict__ deg,
                              float* __restrict__ w) {
  int e = blockIdx.x * blockDim.x + threadIdx.x;
  if (e >= NEDGES) return;
  w[e] = 1.0f / deg[erow[e]];
}

__global__ void inflow_init_kernel(const float* __restrict__ demands,
                                   float* __restrict__ inflow) {
  int n = blockIdx.x * blockDim.x + threadIdx.x;
  if (n >= NNODES) return;
  float d = demands[n];
  inflow[n] = d > 0.0f ? d : 0.0f;
}
__global__ void inflow_acc_kernel(const int* __restrict__ ecol,
                                  const float* __restrict__ flow,
                                  float* __restrict__ inflow) {
  int e = blockIdx.x * blockDim.x + threadIdx.x;
  if (e >= NEDGES) return;
  atomicAdd(&inflow[ecol[e]], flow[e]);
}
__global__ void flow_update_kernel(const int* __restrict__ erow,
                                   const float* __restrict__ w,
                                   const float* __restrict__ inflow,
                                   float* __restrict__ flow) {
  int e = blockIdx.x * blockDim.x + threadIdx.x;
  if (e >= NEDGES) return;
  flow[e] = w[e] * inflow[erow[e]];
}

// ---------------------------------------------------------------------------
extern "C" void kernel_launch(void* const* d_in, const int* in_sizes, int n_in,
                              void* d_out, int out_size, void* d_ws, size_t ws_size,
                              hipStream_t stream) {
  (void)in_sizes; (void)n_in; (void)out_size; (void)ws_size;
  const float* node_emb = (const float*)d_in[0];
  const float* demands  = (const float*)d_in[1];
  const int*   erow     = (const int*)d_in[2];
  const int*   ecol     = (const int*)d_in[3];
  const float* W_gat    = (const float*)d_in[4];
  const float* a_src    = (const float*)d_in[5];
  const float* a_dst    = (const float*)d_in[6];
  const float* Wx       = (const float*)d_in[7];
  const float* Wh       = (const float*)d_in[8];
  const float* b_gru    = (const float*)d_in[9];
  // d_in[10] (Wd) / d_in[11] (bd) are mathematically dead: the per-row softmax
  // of a constant-per-segment logit is exactly 1/degree.

  float* ws = (float*)d_ws;
  size_t off = 0;
  auto alloc = [&](size_t n) { float* p = ws + off; off += n; return p; };
  float* h_cur  = alloc((size_t)NNODES * DIM);
  float* f      = alloc((size_t)HEADS * NNODES * DIM);
  float* ssrc   = alloc((size_t)HEADS * NNODES);
  float* sdst   = alloc((size_t)HEADS * NNODES);
  float* segmax = alloc((size_t)HEADS * NNODES);
  float* segsum = alloc((size_t)HEADS * NNODES);
  float* elog   = alloc((size_t)HEADS * NEDGES);
  float* gat    = alloc((size_t)NNODES * DIM);
  float* G1     = alloc((size_t)NNODES * 3 * DIM);
  float* G2     = alloc((size_t)NNODES * 2 * DIM);
  float* zbuf   = alloc((size_t)NNODES * DIM);
  float* rh     = alloc((size_t)NNODES * DIM);
  float* C3     = alloc((size_t)NNODES * DIM);
  float* deg    = alloc((size_t)NNODES);
  float* wgt    = alloc((size_t)NEDGES);
  float* inflow = alloc((size_t)NNODES);

  float* flow = (float*)d_out;

  hipMemcpyAsync(h_cur, node_emb, sizeof(float) * NNODES * DIM,
                 hipMemcpyDeviceToDevice, stream);

  const int TPB = 256;
  const int GXB = (NNODES / 16 + (GTPB / 32) - 1) / (GTPB / 32);  // row-strip blocks
  auto b256 = [](long n) { return (int)((n + 255) / 256); };

  for (int layer = 0; layer < GLAYERS; ++layer) {
    // f[h] = h @ W_gat[h]  (4 batched 20000x64x64 f32 WMMA GEMMs)
    gemm_wmma_kernel<<<dim3(GXB, HEADS, 1), GTPB, 0, stream>>>(
        h_cur, DIM, 0, W_gat, DIM, (long)DIM * DIM,
        f, DIM, (long)NNODES * DIM, NNODES, nullptr);

    sdot_kernel<<<(HEADS * NNODES + 7) / 8, TPB, 0, stream>>>(f, a_src, a_dst,
                                                              ssrc, sdst);
    fill_kernel<<<b256(HEADS * NNODES), TPB, 0, stream>>>(segmax, -INFINITY,
                                                          HEADS * NNODES);
    fill_kernel<<<b256(HEADS * NNODES), TPB, 0, stream>>>(segsum, 0.0f,
                                                          HEADS * NNODES);
    fill_kernel<<<b256((long)NNODES * DIM), TPB, 0, stream>>>(gat, 0.0f,
                                                              NNODES * DIM);
    edge_logit_kernel<<<b256(NEDGES), TPB, 0, stream>>>(ssrc, sdst, erow, ecol,
                                                        elog, segmax);
    edge_exp_kernel<<<b256(NEDGES), TPB, 0, stream>>>(erow, elog, segmax, segsum);
    scatter_kernel<<<(NEDGES + 7) / 8, TPB, 0, stream>>>(f, elog, segsum,
                                                         erow, ecol, gat);

    // G1 = gat @ Wx + b_gru  [N,192] (3 column panels)
    gemm_wmma_kernel<<<dim3(GXB, 1, 3), GTPB, 0, stream>>>(
        gat, DIM, 0, Wx, 3 * DIM, 0, G1, 3 * DIM, 0, NNODES, b_gru);
    // G2 = h @ Wh[:, :128]  [N,128] (2 column panels)
    gemm_wmma_kernel<<<dim3(GXB, 1, 2), GTPB, 0, stream>>>(
        h_cur, DIM, 0, Wh, 3 * DIM, 0, G2, 2 * DIM, 0, NNODES, nullptr);

    gru1_kernel<<<b256((long)NNODES * DIM), TPB, 0, stream>>>(G1, G2, h_cur,
                                                              zbuf, rh);
    // C3 = (r*h) @ Wh[:, 128:192]
    gemm_wmma_kernel<<<dim3(GXB, 1, 1), GTPB, 0, stream>>>(
        rh, DIM, 0, Wh + 2 * DIM, 3 * DIM, 0, C3, DIM, 0, NNODES, nullptr);

    gru2_kernel<<<b256((long)NNODES * DIM), TPB, 0, stream>>>(G1, C3, zbuf, h_cur);
  }

  // edge weights = 1/degree (exact seg_softmax of constant-per-row logits)
  fill_kernel<<<b256(NNODES), TPB, 0, stream>>>(deg, 0.0f, NNODES);
  degree_kernel<<<b256(NEDGES), TPB, 0, stream>>>(erow, deg);
  invdeg_kernel<<<b256(NEDGES), TPB, 0, stream>>>(erow, deg, wgt);

  // iterative sparse MCF flow
  fill_kernel<<<b256(NEDGES), TPB, 0, stream>>>(flow, 0.0f, NEDGES);
  for (int it = 0; it < FLOWITERS; ++it) {
    inflow_init_kernel<<<b256(NNODES), TPB, 0, stream>>>(demands, inflow);
    inflow_acc_kernel<<<b256(NEDGES), TPB, 0, stream>>>(ecol, flow, inflow);
    flow_update_kernel<<<b256(NEDGES), TPB, 0, stream>>>(erow, wgt, inflow, flow);
  }
}